// DeepBidirectionalRNN_27857157882244
// MI455X (gfx1250) — compile-verified
//
#include <hip/hip_runtime.h>
#include <hip/hip_bf16.h>
#include <math.h>

// ---------------------------------------------------------------------------
// DeepBidirectionalRNN for MI455X (gfx1250).
//   T=8192, IDIM=512, HDIM=1024, NL=3, ODIM=1024, all fp32.
//
// d_in index map (assumes JAX tree_flatten => sorted dict keys):
//   {'params': {'Wout','bout','layers':[{'bw':{'Whh','Wih','b'},'fw':{...}} x3]}, 'x': x}
//   0: Wout (1024x2048)       1: bout (1024)
//   layer l (l=0..2), base = 2 + 6*l:
//     +0 bw.Whh (1024x1024)  +1 bw.Wih (1024 x in_dim)  +2 bw.b (1024)
//     +3 fw.Whh (1024x1024)  +4 fw.Wih (1024 x in_dim)  +5 fw.b (1024)
//   20: x (8192x512)
//
// d_out layout (floats): [ out(1024) | l0 h_fw | l0 h_bw | l1 h_fw | l1 h_bw | l2 h_fw | l2 h_bw ]
// ---------------------------------------------------------------------------

#define RNN_T    8192
#define RNN_IDIM 512
#define RNN_H    1024
#define RNN_NL   3

typedef __attribute__((ext_vector_type(2))) float v2f;
typedef __attribute__((ext_vector_type(8))) float v8f;

// ---------------------------------------------------------------------------
// pre = X @ Wih^T + b via V_WMMA_F32_16X16X4_F32 (native fp32 matrix op).
// X row t is split across two buffers (x0|x1), each of width w0, implementing
// concat([h_fw, h_bw], axis=1) for layers 1..2 without materializing concat.
// The k-loop is split into an x0 phase and an x1 phase so the hot loop has no
// per-iteration select: just pointer-bump + global_load_b64 (+imm offsets
// after unroll) + v_wmma. One wave owns one 16x16 tile of pre; the 8 waves of
// a block share the same 16 x-rows (L0 reuse across nCol tiles).
// ---------------------------------------------------------------------------
__global__ __launch_bounds__(256) void rnn_gemm_pre_kernel(
    const float* __restrict__ x0, const float* __restrict__ x1, int w0, int K,
    const float* __restrict__ Wih, const float* __restrict__ bias,
    float* __restrict__ pre)
{
    const int wave = threadIdx.x >> 5;
    const int lane = threadIdx.x & 31;
    const int half = lane >> 4;     // 0: lanes 0-15, 1: lanes 16-31
    const int r    = lane & 15;
    const int kh   = half << 1;     // this half's K sub-offset {0,2}

    // tile id -> (t-tile, n-tile); consecutive waves share the same t rows.
    const int tileId = blockIdx.x * 8 + wave;
    const int tRow = tileId >> 6;            // / (HDIM/16 = 64)
    const int nCol = tileId & 63;
    const int tBase = tRow << 4;
    const int nBase = nCol << 4;
    const int xt    = tBase + r;             // A row

    // B row pointer (= Wih^T column), advanced through both phases.
    const float* wp = Wih + (size_t)(nBase + r) * K + kh;

    v8f acc = {0.f, 0.f, 0.f, 0.f, 0.f, 0.f, 0.f, 0.f};

    // ---- phase 0: k in [0, w0) from x0 ----
    {
        const float* xp = x0 + (size_t)xt * w0 + kh;
#pragma unroll 8
        for (int kk = 0; kk < w0; kk += 4) {
            v2f a, b;
            a.x = xp[0]; a.y = xp[1];        // A: K = kh, kh+1
            b.x = wp[0]; b.y = wp[1];        // B: same K, column nBase+r
            // (neg_a, A, neg_b, B, c_mod, C, reuse_a, reuse_b)
            acc = __builtin_amdgcn_wmma_f32_16x16x4_f32(
                false, a, false, b, (short)0, acc, false, false);
            xp += 4; wp += 4;
        }
    }

    // ---- phase 1: k in [w0, 2*w0) from x1 (uniform branch) ----
    if (x1 != nullptr) {
        const float* xp = x1 + (size_t)xt * w0 + kh;
#pragma unroll 8
        for (int kk = 0; kk < w0; kk += 4) {
            v2f a, b;
            a.x = xp[0]; a.y = xp[1];
            b.x = wp[0]; b.y = wp[1];
            acc = __builtin_amdgcn_wmma_f32_16x16x4_f32(
                false, a, false, b, (short)0, acc, false, false);
            xp += 4; wp += 4;
        }
    }

    const float bj = bias[nBase + r];
#pragma unroll
    for (int v = 0; v < 8; ++v) {
        // D layout: VGPR v -> row (v + 8*half), col r
        pre[(size_t)(tBase + v + (half << 3)) * RNN_H + nBase + r] = acc[v] + bj;
    }
}

// ---------------------------------------------------------------------------
// Whh (1024x1024) -> Whh^T so the scan's per-step weight reads are coalesced.
// ---------------------------------------------------------------------------
__global__ __launch_bounds__(256) void rnn_transpose_kernel(
    const float* __restrict__ A, float* __restrict__ At)
{
    __shared__ float tile[32][33];
    const int bx = blockIdx.x * 32, by = blockIdx.y * 32;
    const int x = threadIdx.x;
#pragma unroll
    for (int yy = threadIdx.y; yy < 32; yy += 8)
        tile[yy][x] = A[(size_t)(by + yy) * RNN_H + bx + x];
    __syncthreads();
#pragma unroll
    for (int yy = threadIdx.y; yy < 32; yy += 8)
        At[(size_t)(bx + yy) * RNN_H + by + x] = tile[x][yy];
}

// ---------------------------------------------------------------------------
// Sequential recurrence, one block per direction (blockIdx.x: 0=fw, 1=bw).
// H block initially holds pre[t]; we overwrite it in place with h[t].
// h_t = tanh(pre_t + Whh h_{t-1}); h lives in LDS (lane-uniform broadcast),
// weights stream as float4 (256 thr x 16B = one full 4KB Whh^T row per k),
// L2-resident (6 x 4MB << 192MB L2).
// ---------------------------------------------------------------------------
__global__ __launch_bounds__(256) void rnn_scan_kernel(
    const float* __restrict__ WTfw, const float* __restrict__ WTbw,
    float* __restrict__ Hfw, float* __restrict__ Hbw)
{
    const int dir = blockIdx.x;
    const float* WT = dir ? WTbw : WTfw;
    float*       Hb = dir ? Hbw  : Hfw;

    __shared__ float hs[RNN_H];
    const int tid = threadIdx.x;        // 0..255
    const int j4  = tid << 2;           // 4 outputs per thread
    hs[j4 + 0] = 0.f; hs[j4 + 1] = 0.f; hs[j4 + 2] = 0.f; hs[j4 + 3] = 0.f;
    __syncthreads();

    for (int step = 0; step < RNN_T; ++step) {
        const int t  = dir ? (RNN_T - 1 - step) : step;
        const int tn = dir ? (t > 0 ? t - 1 : t) : (t + 1 < RNN_T ? t + 1 : t);
        __builtin_prefetch(Hb + (size_t)tn * RNN_H + j4, 0, 0); // global_prefetch_b8

        const float4 p = *(const float4*)(Hb + (size_t)t * RNN_H + j4);
        float a0 = p.x, a1 = p.y, a2 = p.z, a3 = p.w;

#pragma unroll 8
        for (int k = 0; k < RNN_H; ++k) {
            const float  hk = hs[k];                         // LDS broadcast
            const float4 w  = *(const float4*)(WT + (size_t)k * RNN_H + j4);
            a0 += w.x * hk; a1 += w.y * hk; a2 += w.z * hk; a3 += w.w * hk;
        }
        const float h0 = tanhf(a0), h1 = tanhf(a1), h2 = tanhf(a2), h3 = tanhf(a3);

        __syncthreads();                 // all reads of hs done
        hs[j4 + 0] = h0; hs[j4 + 1] = h1; hs[j4 + 2] = h2; hs[j4 + 3] = h3;
        float4 o; o.x = h0; o.y = h1; o.z = h2; o.w = h3;
        *(float4*)(Hb + (size_t)t * RNN_H + j4) = o;
        __syncthreads();                 // hs fully updated for next step
    }
}

// ---------------------------------------------------------------------------
// out = Wout @ concat(h_fw[T-1], h_bw[0]) + bout
// ---------------------------------------------------------------------------
__global__ __launch_bounds__(1024) void rnn_out_kernel(
    const float* __restrict__ Wout, const float* __restrict__ bout,
    const float* __restrict__ Hfw,  const float* __restrict__ Hbw,
    float* __restrict__ out)
{
    __shared__ float feat[2 * RNN_H];
    const int tid = threadIdx.x;
    feat[tid]         = Hfw[(size_t)(RNN_T - 1) * RNN_H + tid];
    feat[tid + RNN_H] = Hbw[tid];
    __syncthreads();

    float acc = bout[tid];
    const float* wr = Wout + (size_t)tid * (2 * RNN_H);
#pragma unroll 4
    for (int k = 0; k < 2 * RNN_H; ++k) acc += wr[k] * feat[k];
    out[tid] = acc;
}

// ---------------------------------------------------------------------------
extern "C" void kernel_launch(void* const* d_in, const int* in_sizes, int n_in,
                              void* d_out, int out_size, void* d_ws, size_t ws_size,
                              hipStream_t stream)
{
    (void)in_sizes; (void)n_in; (void)out_size; (void)ws_size;

    const float* Wout = (const float*)d_in[0];
    const float* bout = (const float*)d_in[1];
    const float* x    = (const float*)d_in[2 + 6 * RNN_NL];   // index 20

    float* out   = (float*)d_out;
    float* hbase = (float*)d_out + 1024;
    float* WTfw  = (float*)d_ws;                              // 4 MB
    float* WTbw  = WTfw + (size_t)RNN_H * RNN_H;              // 4 MB

    const int tiles  = (RNN_T / 16) * (RNN_H / 16);           // 32768
    const int blocks = tiles / 8;

    for (int l = 0; l < RNN_NL; ++l) {
        const int base = 2 + 6 * l;
        const float* WhhBw = (const float*)d_in[base + 0];
        const float* WihBw = (const float*)d_in[base + 1];
        const float* bBw   = (const float*)d_in[base + 2];
        const float* WhhFw = (const float*)d_in[base + 3];
        const float* WihFw = (const float*)d_in[base + 4];
        const float* bFw   = (const float*)d_in[base + 5];

        float* Hfw = hbase + (size_t)l * 2 * RNN_T * RNN_H;
        float* Hbw = Hfw + (size_t)RNN_T * RNN_H;

        dim3 tb(32, 8), tg(RNN_H / 32, RNN_H / 32);
        rnn_transpose_kernel<<<tg, tb, 0, stream>>>(WhhFw, WTfw);
        rnn_transpose_kernel<<<tg, tb, 0, stream>>>(WhhBw, WTbw);

        const float* x0; const float* x1; int w0, K;
        if (l == 0) { x0 = x; x1 = nullptr; w0 = RNN_IDIM; K = RNN_IDIM; }
        else {
            x0 = Hfw - (size_t)2 * RNN_T * RNN_H;   // previous layer h_fw
            x1 = x0 + (size_t)RNN_T * RNN_H;        // previous layer h_bw
            w0 = RNN_H; K = 2 * RNN_H;
        }

        // pre written straight into the h output blocks (scan rewrites in place)
        rnn_gemm_pre_kernel<<<blocks, 256, 0, stream>>>(x0, x1, w0, K, WihFw, bFw, Hfw);
        rnn_gemm_pre_kernel<<<blocks, 256, 0, stream>>>(x0, x1, w0, K, WihBw, bBw, Hbw);

        rnn_scan_kernel<<<2, 256, 0, stream>>>(WTfw, WTbw, Hfw, Hbw);
    }

    const float* Hfw2 = hbase + (size_t)(RNN_NL - 1) * 2 * RNN_T * RNN_H;
    const float* Hbw2 = Hfw2 + (size_t)RNN_T * RNN_H;
    rnn_out_kernel<<<1, 1024, 0, stream>>>(Wout, bout, Hfw2, Hbw2, out);
}